// ROIArea_80547816669749
// MI455X (gfx1250) — compile-verified
//
#include <hip/hip_runtime.h>

typedef __attribute__((ext_vector_type(2))) float v2f;
typedef __attribute__((ext_vector_type(8))) float v8f;

#define C_CH   256
#define H_IN   200
#define W_IN   200
#define HW_IN  (H_IN * W_IN)
#define K_ROI  512
#define OH     7
#define OW     7
#define SCALE  0.25f

// One block per ROI, 256 threads = 8 wave32s.
// Y-interpolation runs on V_WMMA_F32_16X16X4_F32:
//   D(16x16: rows=oh, cols=16 channels) = Wy(7x14 padded) @ H(14 x 16 channels)
// chained as 4 K=4 chunks. X-interpolation feeds the B fragments from
// L2-resident gathers (entire input = 163.8 MB fits in the 192 MB L2).
// Loads are batched per-ow and software-pipelined one ow ahead so the 16-load
// clause of ow+1 is in flight while the WMMA chain of ow executes.
__global__ __launch_bounds__(256, 4) void roi_area_wmma_kernel(
    const float* __restrict__ inputs,   // (4, 256, 200, 200)
    const float* __restrict__ rois,     // (512, 5)
    const float* __restrict__ masks,    // (512, 7, 7)
    float* __restrict__ out)            // (512, 256, 7, 7)
{
  __shared__ int   s_yidx[16];   // corner rows: [2*oh]=y0, [2*oh+1]=y1 (14 used)
  __shared__ float s_wy[16];     // y weights:   [2*oh]=1-ly, [2*oh+1]=ly
  __shared__ int   s_xb0[8];     // x0 * 4 (byte offset)
  __shared__ int   s_xb1[8];     // x1 * 4 (byte offset)
  __shared__ float s_lx[8];
  __shared__ float s_mask[49];
  __shared__ int   s_b;

  const int k = blockIdx.x;
  const int t = threadIdx.x;

  if (t == 0) s_b = (int)rois[(size_t)k * 5 + 0];

  if (t < 7) {
    const float rx1 = rois[(size_t)k*5 + 1] * SCALE;
    const float ry1 = rois[(size_t)k*5 + 2] * SCALE;
    const float rx2 = rois[(size_t)k*5 + 3] * SCALE;
    const float ry2 = rois[(size_t)k*5 + 4] * SCALE;
    const float roi_w = fmaxf(rx2 - rx1, 1.0f);
    const float roi_h = fmaxf(ry2 - ry1, 1.0f);
    const float bw = roi_w * (1.0f / OW);
    const float bh = roi_h * (1.0f / OH);

    // y geometry for oh = t
    const float ys  = ry1 + ((float)t + 0.5f) * bh;
    const float y0f = floorf(ys);
    const float ly  = ys - y0f;
    int y0 = (int)y0f;
    y0 = y0 < 0 ? 0 : (y0 > H_IN - 1 ? H_IN - 1 : y0);
    const int y1i = (y0 + 1 > H_IN - 1) ? H_IN - 1 : y0 + 1;
    s_yidx[2*t]     = y0;
    s_yidx[2*t + 1] = y1i;
    s_wy[2*t]       = 1.0f - ly;
    s_wy[2*t + 1]   = ly;

    // x geometry for ow = t
    const float xs  = rx1 + ((float)t + 0.5f) * bw;
    const float x0f = floorf(xs);
    const float lx  = xs - x0f;
    int x0 = (int)x0f;
    x0 = x0 < 0 ? 0 : (x0 > W_IN - 1 ? W_IN - 1 : x0);
    const int x1 = (x0 + 1 > W_IN - 1) ? W_IN - 1 : x0 + 1;
    s_xb0[t] = x0 * 4;
    s_xb1[t] = x1 * 4;
    s_lx[t]  = lx;
  }
  if (t < 49) s_mask[t] = masks[(size_t)k * 49 + t];
  __syncthreads();

  const int wave = t >> 5;
  const int lane = t & 31;
  const int Mrow = lane & 15;
  const int kb   = (lane < 16) ? 0 : 2;   // K striping: V0 -> K+kb, V1 -> K+kb+1

  // A fragments: Wy (7x14 padded to 16x16), 4 chunks of K=4.
  // Wy[oh, r] nonzero only at r=2*oh (1-ly), r=2*oh+1 (ly).
  // Padded rows r=14,15 clamp to row 13 (zero A weight, finite values).
  float aw[4][2];
  int   rcl[8];
#pragma unroll
  for (int kk = 0; kk < 4; ++kk) {
#pragma unroll
    for (int v = 0; v < 2; ++v) {
      const int r = 4*kk + kb + v;
      aw[kk][v]    = (Mrow < 7 && (r >> 1) == Mrow) ? s_wy[r] : 0.0f;
      rcl[kk*2+v]  = (r < 14) ? r : 13;
    }
  }

  // Scalar batch base; all load addresses are (scalar base + zext(32-bit off))
  // so the backend keeps saddr-form global loads with v_add_nc_u32 addressing.
  const int b = __builtin_amdgcn_readfirstlane(s_b);
  const char* base = (const char*)(inputs + (size_t)b * (C_CH * HW_IN));

  for (int g = wave; g < C_CH / 16; g += 8) {
    const int c = g * 16 + Mrow;              // B column N = lane&15 -> channel
    const unsigned cbase = (unsigned)c * HW_IN;

    unsigned rowb[8];                          // per-lane row byte offsets
#pragma unroll
    for (int i = 0; i < 8; ++i)
      rowb[i] = (cbase + (unsigned)s_yidx[rcl[i]] * W_IN) * 4u;

    float* optr = out + ((size_t)k * C_CH + c) * (OH * OW);

    // --- batched gather for one ow: 16 loads, one clause ---
    float q0[8], q1[8];
    {
      const unsigned xb0 = (unsigned)s_xb0[0];
      const unsigned xb1 = (unsigned)s_xb1[0];
#pragma unroll
      for (int i = 0; i < 8; ++i) {
        q0[i] = *(const float*)(base + (rowb[i] + xb0));
        q1[i] = *(const float*)(base + (rowb[i] + xb1));
      }
    }

#pragma unroll
    for (int ow = 0; ow < OW; ++ow) {
      // issue next ow's 16-load clause before consuming current values
      float n0[8], n1[8];
      if (ow + 1 < OW) {
        const unsigned xb0 = (unsigned)s_xb0[ow + 1];
        const unsigned xb1 = (unsigned)s_xb1[ow + 1];
#pragma unroll
        for (int i = 0; i < 8; ++i) {
          n0[i] = *(const float*)(base + (rowb[i] + xb0));
          n1[i] = *(const float*)(base + (rowb[i] + xb1));
        }
      }

      const float lxv = s_lx[ow];
      const float wx0 = 1.0f - lxv;

      v8f acc = {};
#pragma unroll
      for (int kk = 0; kk < 4; ++kk) {
        v2f bfrag, afrag;
        bfrag[0] = fmaf(wx0, q0[kk*2 + 0], lxv * q1[kk*2 + 0]);
        bfrag[1] = fmaf(wx0, q0[kk*2 + 1], lxv * q1[kk*2 + 1]);
        afrag[0] = aw[kk][0];
        afrag[1] = aw[kk][1];
        // D += Wy_chunk(16x4) @ H_chunk(4x16)
        acc = __builtin_amdgcn_wmma_f32_16x16x4_f32(
            false, afrag, false, bfrag, (short)0, acc, false, false);
      }

      // D rows: lanes 0-15 hold M = v (oh 0..6); lanes 16-31 hold padding rows.
      if (lane < 16) {
#pragma unroll
        for (int v = 0; v < 7; ++v) {
          optr[v * OW + ow] = acc[v] * s_mask[v * OW + ow];
        }
      }

      if (ow + 1 < OW) {
#pragma unroll
        for (int i = 0; i < 8; ++i) { q0[i] = n0[i]; q1[i] = n1[i]; }
      }
    }
  }
}

extern "C" void kernel_launch(void* const* d_in, const int* in_sizes, int n_in,
                              void* d_out, int out_size, void* d_ws, size_t ws_size,
                              hipStream_t stream) {
  (void)in_sizes; (void)n_in; (void)out_size; (void)d_ws; (void)ws_size;
  const float* inputs = (const float*)d_in[0];
  const float* rois   = (const float*)d_in[1];
  const float* masks  = (const float*)d_in[2];
  float* out = (float*)d_out;

  roi_area_wmma_kernel<<<K_ROI, 256, 0, stream>>>(inputs, rois, masks, out);
}